// HetConv2d_11244224381462
// MI455X (gfx1250) — compile-verified
//
#include <hip/hip_runtime.h>

typedef __attribute__((ext_vector_type(2))) float v2f;
typedef __attribute__((ext_vector_type(8))) float v8f;

#define CIN   64
#define IMG   256
#define TSX   16      // tile width (pixels)
#define TSY   8       // tile height (pixels)
#define HROW  10      // LDS rows  (TSY + 2 halo)
#define HCOL  18      // valid cols (TSX + 2 halo)
#define RP    21      // padded LDS row pitch (bank-conflict-free: 2*10*21%64=36, 8*10*21%64=16)

__device__ __forceinline__ int lidx(int c, int py, int px) {
  return (c * HROW + py) * RP + px;
}

__global__ __launch_bounds__(128)
void hetconv_wmma_kernel(const float* __restrict__ x,
                         const float* __restrict__ w3,
                         const float* __restrict__ w1,
                         const float* __restrict__ bias,
                         float* __restrict__ y) {
  __shared__ float xs[CIN * HROW * RP];   // 53,760 B

  const int bid = blockIdx.x;
  const int tx = bid & 15;                // 16 tiles across
  const int ty = (bid >> 4) & 31;         // 32 tiles down
  const int b  = bid >> 9;                // batch
  const int x0 = tx * TSX, y0 = ty * TSY;

  const int tid  = threadIdx.x;           // 0..127
  const int lane = tid & 31;
  const int mt   = tid >> 5;              // wave id == M-tile (16 out channels)
  const int hi   = lane >> 4;
  const int lp   = lane & 15;

  // ---------------- fill LDS input tile (zero-padded halo) ----------------
  const float* xb = x + (size_t)b * CIN * IMG * IMG;
  for (int idx = tid; idx < CIN * HROW * HCOL; idx += 128) {   // 11520/128 = 90 exact
    int c  = idx / (HROW * HCOL);
    int r  = idx - c * (HROW * HCOL);
    int ry = r / HCOL;
    int rx = r - ry * HCOL;
    int gy = y0 + ry - 1, gx = x0 + rx - 1;
    float v = 0.0f;
    if (gy >= 0 && gy < IMG && gx >= 0 && gx < IMG)
      v = xb[(c * IMG + gy) * IMG + gx];
    xs[lidx(c, ry, rx)] = v;
  }
  __syncthreads();

  // ---- A operand: fused center-tap 64x64 matrix, resident in registers ----
  // A layout (16x4 f32): lane lp holds row M=lp; vgpr .x -> K=2*hi, .y -> K=2*hi+1
  const int oa = mt * 16 + lp;
  v2f a[16];
#pragma unroll
  for (int k = 0; k < 16; ++k) {
    int c0 = 4 * k + 2 * hi;
    int c1 = c0 + 1;
    float e0 = (((c0 - oa) & 31) == 0) ? w3[(oa * CIN + c0) * 9 + 4] : w1[oa * CIN + c0];
    float e1 = (((c1 - oa) & 31) == 0) ? w3[(oa * CIN + c1) * 9 + 4] : w1[oa * CIN + c1];
    a[k].x = e0;
    a[k].y = e1;
  }

  // bias for the 8 output rows this lane accumulates: o = mt*16 + v + 8*hi
  float bl[8];
#pragma unroll
  for (int v = 0; v < 8; ++v) bl[v] = bias[mt * 16 + 8 * hi + v];

  // ---------------- dense part: 8 N-groups x 16 K-steps of WMMA ----------------
  v8f acc[8];
#pragma unroll
  for (int g = 0; g < 8; ++g) {
    v8f ci;
#pragma unroll
    for (int v = 0; v < 8; ++v) ci[v] = bl[v];
#pragma unroll
    for (int k = 0; k < 16; ++k) {
      // B layout (4x16 f32): lane lp holds col N=lp; vgpr .x -> K=2*hi, .y -> K=2*hi+1
      v2f bb;
      bb.x = xs[lidx(4 * k + 2 * hi,     g + 1, 1 + lp)];
      bb.y = xs[lidx(4 * k + 2 * hi + 1, g + 1, 1 + lp)];
      ci = __builtin_amdgcn_wmma_f32_16x16x4_f32(false, a[k], false, bb,
                                                 (short)0, ci, false, false);
    }
    acc[g] = ci;
  }

  // ---------------- sparse 3x3 taps (2 channels per output, center excluded) ----
#pragma unroll
  for (int v = 0; v < 8; ++v) {
    const int o = mt * 16 + v + 8 * hi;
#pragma unroll
    for (int s = 0; s < 2; ++s) {
      const int c = (o & 31) + 32 * s;
      const float* wr = w3 + ((size_t)o * CIN + c) * 9;
      const float w00 = wr[0], w01 = wr[1], w02 = wr[2];
      const float w10 = wr[3],             w12 = wr[5];
      const float w20 = wr[6], w21 = wr[7], w22 = wr[8];
      float xr[3][3];
#pragma unroll
      for (int d = 0; d < 2; ++d) {
        xr[d][0] = xs[lidx(c, d, lp)];
        xr[d][1] = xs[lidx(c, d, lp + 1)];
        xr[d][2] = xs[lidx(c, d, lp + 2)];
      }
#pragma unroll
      for (int g = 0; g < 8; ++g) {
        const int rn = (g + 2) % 3;           // rolling row window
        xr[rn][0] = xs[lidx(c, g + 2, lp)];
        xr[rn][1] = xs[lidx(c, g + 2, lp + 1)];
        xr[rn][2] = xs[lidx(c, g + 2, lp + 2)];
        const int r0 = g % 3, r1 = (g + 1) % 3;
        float t = acc[g][v];
        t = fmaf(w00, xr[r0][0], t);
        t = fmaf(w01, xr[r0][1], t);
        t = fmaf(w02, xr[r0][2], t);
        t = fmaf(w10, xr[r1][0], t);
        t = fmaf(w12, xr[r1][2], t);
        t = fmaf(w20, xr[rn][0], t);
        t = fmaf(w21, xr[rn][1], t);
        t = fmaf(w22, xr[rn][2], t);
        acc[g][v] = t;
      }
    }
  }

  // ---------------- store (coalesced 64B per half-wave) ----------------
  float* yb = y + (size_t)b * CIN * IMG * IMG;
#pragma unroll
  for (int g = 0; g < 8; ++g) {
#pragma unroll
    for (int v = 0; v < 8; ++v) {
      const int o = mt * 16 + v + 8 * hi;
      yb[(o * IMG + (y0 + g)) * IMG + (x0 + lp)] = acc[g][v];
    }
  }
}

extern "C" void kernel_launch(void* const* d_in, const int* in_sizes, int n_in,
                              void* d_out, int out_size, void* d_ws, size_t ws_size,
                              hipStream_t stream) {
  const float* x  = (const float*)d_in[0];
  const float* w3 = (const float*)d_in[1];
  const float* w1 = (const float*)d_in[2];
  const float* b  = (const float*)d_in[3];
  float* y = (float*)d_out;

  dim3 grid(16 * 32 * 16);   // batch(16) * tilesY(32) * tilesX(16)
  dim3 block(128);           // 4 wave32 -> 4 M-tiles of 16 output channels
  hetconv_wmma_kernel<<<grid, block, 0, stream>>>(x, w3, w1, b, y);
}